// VectorQuantizerEMA_30227979829912
// MI455X (gfx1250) — compile-verified
//
#include <hip/hip_runtime.h>
#include <hip/hip_bf16.h>

// ---------------------------------------------------------------------------
// VQ-VAE EMA vector quantizer for MI455X (gfx1250, wave32).
// Distance GEMM (N=262144, K=1024, D=64, fp32) runs on V_WMMA_F32_16X16X4_F32
// with the full packed codebook DMA'd into 260KB of LDS by the Tensor Data
// Mover (tensor_load_to_lds + s_wait_tensorcnt). Segment sums accumulate in
// LDS (ds_add_f32) before a single global flush.
// ---------------------------------------------------------------------------

typedef __attribute__((ext_vector_type(2))) float        v2f;
typedef __attribute__((ext_vector_type(8))) float        v8f;
typedef __attribute__((ext_vector_type(4))) unsigned int v4u;
typedef __attribute__((ext_vector_type(8))) int          v8i;
typedef __attribute__((ext_vector_type(4))) int          v4i;

#define NVEC   262144      // B*H*W vectors
#define DDIM   64
#define KCODE  1024
#define HW     4096        // H*W
#define CHW    262144      // C*H*W
#define NQELT  16777216    // B*C*H*W

// workspace layout (float offsets)
#define WS_PACKB   0          // 65536  packed WMMA B-layout codebook
#define WS_BIAS    65536      // 1024   0.5*||w_k||^2
#define WS_COUNTS  66560      // 1024   segment counts
#define WS_DW      67584      // 65536  segment sums  (counts+dw zeroed together)
#define WS_CSRAW   133120     // 1024   cs before smoothing
#define WS_NTOT    134144     // 1      sum(cs)
#define WS_NEWW    134208     // 65536  updated codebook

// ---------------------------------------------------------------------------
__global__ void k_init(float* __restrict__ ws, float* __restrict__ out) {
    int t = blockIdx.x * blockDim.x + threadIdx.x;
    if (t < 66560) ws[WS_COUNTS + t] = 0.0f;          // counts + dw
    else if (t == 66560) ws[WS_NTOT] = 0.0f;
    else if (t == 66561) out[0] = 0.0f;               // loss accumulator
}

// ---------------------------------------------------------------------------
// Pack weight[K][D] into V_WMMA_F32_16X16X4_F32 B-matrix lane layout:
// per (ktile,kk) the wave's 32 lanes each hold a float2 = w[code][d0], w[code][d0+1]
// with code = ktile*16 + (lane&15), d0 = 4*kk + (lane<16 ? 0 : 2).
__global__ void k_pack(const float* __restrict__ w,
                       float* __restrict__ packB, float* __restrict__ bias) {
    int t = blockIdx.x * blockDim.x + threadIdx.x;
    if (t < 32768) {                      // 64 ktiles * 16 kk * 32 lanes
        int lane = t & 31;
        int kk   = (t >> 5) & 15;
        int kt   = t >> 9;
        int k    = kt * 16 + (lane & 15);
        int d0   = 4 * kk + ((lane < 16) ? 0 : 2);
        packB[2 * t]     = w[k * DDIM + d0];
        packB[2 * t + 1] = w[k * DDIM + d0 + 1];
    } else if (t < 32768 + KCODE) {
        int k = t - 32768;
        float s = 0.0f;
        for (int d = 0; d < DDIM; ++d) { float v = w[k * DDIM + d]; s += v * v; }
        bias[k] = 0.5f * s;
    }
}

// ---------------------------------------------------------------------------
// Build a 1-D TDM descriptor: copy `bytes` (multiple of 8) from `gaddr`
// (global) to LDS byte offset `lds_off`, data_size=8B.  Groups per CDNA5 D#.
// This toolchain carries the 6-arg builtin (extra int32x8 group, zero-filled).
__device__ __forceinline__ void tdm_load_1d(unsigned long long gaddr,
                                            unsigned lds_off, unsigned bytes) {
    unsigned n8 = bytes >> 3;                       // elements of 8 bytes
    v4u g0;
    g0[0] = 1u;                                     // count=1, user mode
    g0[1] = lds_off;                                // lds_addr (bytes)
    g0[2] = (unsigned)(gaddr & 0xFFFFFFFFu);        // global_addr[31:0]
    g0[3] = (unsigned)((gaddr >> 32) & 0x01FFFFFFu) // global_addr[56:32]
          | 0x80000000u;                            // type=2 ("image")
    v8i g1;
    g1[0] = 0x00030000;                             // workgroup_mask=0, data_size=3 (8B)
    g1[1] = (int)(n8 << 16);                        // tensor_dim0[15:0] in bits[63:48]
    g1[2] = (int)(n8 >> 16) | 0x00010000;           // tensor_dim0[31:16]; tensor_dim1=1
    g1[3] = (int)(n8 << 16);                        // tile_dim0 = n8 (bits[127:112])
    g1[4] = 0;                                      // tile_dim1=0 (unused), tile_dim2=0
    g1[5] = (int)n8;                                // tensor_dim0_stride[31:0]
    g1[6] = 0;                                      // tensor_dim0_stride[47:32]=0
    g1[7] = 0;
    v4i gz = {0, 0, 0, 0};                          // groups 2/3 unused (<=2D)
    v8i z8 = {0, 0, 0, 0, 0, 0, 0, 0};              // extra group (clang-23 form)
    __builtin_amdgcn_tensor_load_to_lds(g0, g1, gz, gz, z8, 0);
}

// ---------------------------------------------------------------------------
// Nearest-code search. Block = 256 threads = 8 waves; each wave owns one
// 16-row x tile and scans all 1024 codes as 32 pairs of 16-code tiles with
// two independent WMMA accumulator chains (hides XDL latency).
// Codebook arrives in LDS via the Tensor Data Mover.
__global__ __launch_bounds__(256) void k_assign(const float* __restrict__ x,
                                                const float* __restrict__ packB,
                                                const float* __restrict__ bias,
                                                int* __restrict__ idx_out) {
    extern __shared__ float lds[];
    float* ldsB    = lds;            // 65536 floats (256KB) @ LDS offset 0
    float* ldsBias = lds + 65536;    // 1024 floats         @ LDS offset 262144

    // TDM DMA of packed codebook + bias straight into LDS (no VGPR round-trip).
    if (threadIdx.x == 0) {          // one wave issues; EXEC-independent DMA
        tdm_load_1d((unsigned long long)(uintptr_t)packB, 0u,      262144u);
        tdm_load_1d((unsigned long long)(uintptr_t)bias,  262144u, 4096u);
        __builtin_amdgcn_s_wait_tensorcnt(0);
    }
    __syncthreads();

    const int wave = threadIdx.x >> 5;
    const int lane = threadIdx.x & 31;
    const int tile = blockIdx.x * 8 + wave;

    // A-matrix preload: lane holds row M = lane&15, K-pair per the 16x4 layout
    const int row = (tile << 4) + (lane & 15);
    const float* xrow = x + (size_t)(row >> 12) * CHW + (row & 4095);
    const int off = (lane < 16) ? 0 : 2;
    float areg[32];
    #pragma unroll
    for (int kk = 0; kk < 16; ++kk) {
        int d0 = 4 * kk + off;
        areg[2 * kk]     = xrow[(size_t)d0 * HW];
        areg[2 * kk + 1] = xrow[(size_t)(d0 + 1) * HW];
    }

    float best[8];
    int   bidx[8];
    #pragma unroll
    for (int r = 0; r < 8; ++r) { best[r] = -3.4e38f; bidx[r] = 0; }

    const v2f* Bp = (const v2f*)ldsB;
    for (int kt = 0; kt < 32; ++kt) {
        v8f acc0 = {};
        v8f acc1 = {};
        #pragma unroll
        for (int kk = 0; kk < 16; ++kk) {
            v2f av; av[0] = areg[2 * kk]; av[1] = areg[2 * kk + 1];
            v2f bv0 = Bp[((2 * kt) * 16 + kk) * 32 + lane];
            v2f bv1 = Bp[((2 * kt + 1) * 16 + kk) * 32 + lane];
            acc0 = __builtin_amdgcn_wmma_f32_16x16x4_f32(
                false, av, false, bv0, (short)0, acc0, false, false);
            acc1 = __builtin_amdgcn_wmma_f32_16x16x4_f32(
                false, av, false, bv1, (short)0, acc1, false, false);
        }
        // score = x.w - 0.5||w||^2 ; column (code) = lane&15 for every acc VGPR
        float bc0 = ldsBias[(2 * kt) * 16 + (lane & 15)];
        float bc1 = ldsBias[(2 * kt + 1) * 16 + (lane & 15)];
        int  c0 = (2 * kt) * 16 + (lane & 15);
        int  c1 = c0 + 16;
        #pragma unroll
        for (int r = 0; r < 8; ++r) {           // ascending k keeps argmin tie rule
            float s0 = acc0[r] - bc0;
            if (s0 > best[r]) { best[r] = s0; bidx[r] = c0; }
            float s1 = acc1[r] - bc1;
            if (s1 > best[r]) { best[r] = s1; bidx[r] = c1; }
        }
    }

    // cross-lane argmax inside each 16-lane half (rows r and r+8 per VGPR slot)
    #pragma unroll
    for (int r = 0; r < 8; ++r) {
        float v = best[r]; int id = bidx[r];
        #pragma unroll
        for (int m = 8; m >= 1; m >>= 1) {
            float ov = __shfl_xor(v, m, 16);
            int  oid = __shfl_xor(id, m, 16);
            if (ov > v || (ov == v && oid < id)) { v = ov; id = oid; }
        }
        best[r] = v; bidx[r] = id;
    }
    if ((lane & 15) == 0) {
        int base = (tile << 4) + ((lane >> 4) << 3);
        #pragma unroll
        for (int r = 0; r < 8; ++r) idx_out[base + r] = bidx[r];
    }
}

// ---------------------------------------------------------------------------
// Segment sums via LDS accumulation: the whole [K=1024, D=64] accumulator
// (padded to stride 65 -> bank = (k*65+d)%64 spreads over k) lives in LDS.
// 64 blocks each accumulate 4096 vectors locally, then flush once.
__global__ __launch_bounds__(256) void k_scatter(const float* __restrict__ x,
                                                 const int* __restrict__ idx,
                                                 float* __restrict__ counts,
                                                 float* __restrict__ dw) {
    extern __shared__ float lds[];
    float* ldw  = lds;            // 1024 * 65 floats (padded)
    float* lcnt = lds + 66560;    // 1024 floats

    for (int i = threadIdx.x; i < 66560; i += 256) ldw[i] = 0.0f;
    for (int i = threadIdx.x; i < KCODE; i += 256) lcnt[i] = 0.0f;
    __syncthreads();

    const int base = blockIdx.x * 4096;
    for (int it = 0; it < 16; ++it) {
        int n = base + it * 256 + threadIdx.x;
        int k = idx[n];
        atomicAdd(&lcnt[k], 1.0f);                       // ds_add_f32
        const float* xr = x + (size_t)(n >> 12) * CHW + (n & 4095);
        float* rowp = ldw + k * 65;
        #pragma unroll 8
        for (int d = 0; d < DDIM; ++d) atomicAdd(&rowp[d], xr[(size_t)d * HW]);
    }
    __syncthreads();

    for (int t = threadIdx.x; t < KCODE * DDIM; t += 256) {
        float v = ldw[(t >> 6) * 65 + (t & 63)];
        if (v != 0.0f) atomicAdd(&dw[t], v);
    }
    for (int k = threadIdx.x; k < KCODE; k += 256) {
        float c = lcnt[k];
        if (c != 0.0f) atomicAdd(&counts[k], c);
    }
}

__global__ void k_ema(const float* __restrict__ ema_cs, const float* __restrict__ counts,
                      float* __restrict__ cs_raw, float* __restrict__ ntot) {
    int k = blockIdx.x * blockDim.x + threadIdx.x;
    if (k < KCODE) {
        float cs = ema_cs[k] * 0.99f + 0.01f * counts[k];
        cs_raw[k] = cs;
        atomicAdd(ntot, cs);
    }
}

__global__ void k_neww(const float* __restrict__ ema_w, const float* __restrict__ dw,
                       const float* __restrict__ cs_raw, const float* __restrict__ ntot,
                       float* __restrict__ neww) {
    int t = blockIdx.x * blockDim.x + threadIdx.x;   // 65536 = K*D
    int k = t >> 6;
    float n  = *ntot;
    float cs = cs_raw[k];
    float css = (cs + 1e-5f) / (n + (float)KCODE * 1e-5f) * n;  // Laplace smoothing
    neww[t] = (ema_w[t] * 0.99f + 0.01f * dw[t]) / css;
}

// ---------------------------------------------------------------------------
__global__ void k_quant(const float* __restrict__ x, const int* __restrict__ idx,
                        const float* __restrict__ neww, float* __restrict__ out) {
    int n = blockIdx.x * blockDim.x + threadIdx.x;
    int k = idx[n];
    const float* q = neww + k * DDIM;
    size_t base = (size_t)(n >> 12) * CHW + (n & 4095);
    float loc = 0.0f;
    #pragma unroll 8
    for (int d = 0; d < DDIM; ++d) {
        float xv = x[base + (size_t)d * HW];
        float qv = q[d];
        float df = qv - xv;
        loc += df * df;                               // (quantized - x)^2
        out[1 + base + (size_t)d * HW] = xv + df;     // straight-through value
    }
    #pragma unroll
    for (int m = 16; m >= 1; m >>= 1) loc += __shfl_xor(loc, m, 32);
    if ((threadIdx.x & 31) == 0)
        atomicAdd(&out[0], loc * (0.25f / (float)NQELT));
}

// ---------------------------------------------------------------------------
extern "C" void kernel_launch(void* const* d_in, const int* in_sizes, int n_in,
                              void* d_out, int out_size, void* d_ws, size_t ws_size,
                              hipStream_t stream) {
    const float* x       = (const float*)d_in[0];   // [64,64,64,64] NCHW
    const float* weight  = (const float*)d_in[1];   // [1024,64]
    const float* ema_cs  = (const float*)d_in[2];   // [1024]
    const float* ema_w   = (const float*)d_in[3];   // [1024,64]
    float* out = (float*)d_out;                     // [loss | quantized NCHW | idx]
    float* ws  = (float*)d_ws;
    int*   idxp = (int*)(out + 1 + NQELT);          // idx region (raw int32)

    // 1) zero accumulators
    k_init<<<(66562 + 255) / 256, 256, 0, stream>>>(ws, out);
    // 2) pack codebook into WMMA B-layout + per-code bias
    k_pack<<<(33792 + 255) / 256, 256, 0, stream>>>(weight, ws + WS_PACKB, ws + WS_BIAS);
    // 3) nearest-code search (WMMA f32, codebook TDM-loaded into 260KB LDS)
    k_assign<<<NVEC / 16 / 8, 256, (65536 + 1024) * sizeof(float), stream>>>(
        x, ws + WS_PACKB, ws + WS_BIAS, idxp);
    // 4) segment sums (LDS accumulate, 270KB LDS, then single flush)
    k_scatter<<<NVEC / 4096, 256, (66560 + 1024) * sizeof(float), stream>>>(
        x, idxp, ws + WS_COUNTS, ws + WS_DW);
    // 5) EMA cluster sizes + total
    k_ema<<<KCODE / 256, 256, 0, stream>>>(ema_cs, ws + WS_COUNTS, ws + WS_CSRAW, ws + WS_NTOT);
    // 6) smoothed update -> new codebook
    k_neww<<<(KCODE * DDIM) / 256, 256, 0, stream>>>(ema_w, ws + WS_DW, ws + WS_CSRAW,
                                                     ws + WS_NTOT, ws + WS_NEWW);
    // 7) quantize with updated codebook, straight-through output, loss
    k_quant<<<NVEC / 256, 256, 0, stream>>>(x, idxp, ws + WS_NEWW, out);
}